// Freq_attention_78778290143392
// MI455X (gfx1250) — compile-verified
//
#include <hip/hip_runtime.h>
#include <math.h>

#define N_  16
#define C_  128
#define T_  513
#define TI  257   // i dimension (x1 = x[:,:,0::2])
#define TJ  256   // j dimension (x2 = x[:,:,1::2])
#define TIP 272   // i padded to 17*16 tiles
#define ITILES 17
#define JTILES 16
#define TILE_F 2048   // floats per packed 16-row x 128-K tile
#define EPS_ 1e-5f

typedef __attribute__((ext_vector_type(2))) float v2f;
typedef __attribute__((ext_vector_type(8))) float v8f;

// ---- kernel 1: per-position sum of squares over channels (coalesced over t) ----
// S1[n,i] = sum_c x[n,c,2i]^2 ; S2[n,j] = sum_c x[n,c,2j+1]^2
__global__ void k_prep(const float* __restrict__ x,
                       float* __restrict__ S1, float* __restrict__ S2) {
    const int t = blockIdx.x * blockDim.x + threadIdx.x;
    const int n = blockIdx.y;
    if (t >= T_) return;
    const float* xp = x + (size_t)n * C_ * T_ + t;
    float s = 0.f;
    #pragma unroll 4
    for (int c = 0; c < C_; ++c) {
        const float v = xp[(size_t)c * T_];
        s += v * v;
    }
    if (t & 1) S2[n * TJ  + (t >> 1)] = s;
    else       S1[n * TIP + (t >> 1)] = s;
}

// ---- kernel 2: pack x1/x2 into WMMA-native tiled layout ----
// Per 16-row tile: offset = (kc*32 + half*16 + mrow)*2 + elem
//   kc = c>>2, half = (c>>1)&1, elem = c&1, mrow = row&15
// => GEMM lane l does one b64 load per kc: exactly (K=2*half, K=2*half+1) of row mrow.
__global__ void k_pack(const float* __restrict__ x,
                       float* __restrict__ x1w, float* __restrict__ x2w) {
    const size_t idx = (size_t)blockIdx.x * blockDim.x + threadIdx.x;
    if (idx >= (size_t)N_ * C_ * T_) return;
    const int n = (int)(idx / ((size_t)C_ * T_));
    const int r = (int)(idx % ((size_t)C_ * T_));
    const int c = r / T_;
    const int t = r % T_;
    const float v = x[idx];
    const int pos  = t >> 1;                 // i or j
    const int tile = pos >> 4;
    const int lane = ((c >> 1) & 1) * 16 + (pos & 15);   // half*16 + mrow
    const size_t off = ((size_t)(c >> 2) * 32 + lane) * 2 + (c & 1);
    if (t & 1) x2w[((size_t)n * JTILES + tile) * TILE_F + off] = v;
    else       x1w[((size_t)n * ITILES + tile) * TILE_F + off] = v;
}

// ---- kernel 3: batched GEMM via WMMA + fused exp/threshold ----
// One wave per 16x16 output tile; K=128 = 32 x v_wmma_f32_16x16x4_f32.
// Inner loop: 2 coalesced global_load_b64 + 1 WMMA, zero divergence.
__global__ __launch_bounds__(32) void k_gemm(const float* __restrict__ x1w,
                                             const float* __restrict__ x2w,
                                             const float* __restrict__ S1,
                                             const float* __restrict__ S2,
                                             float* __restrict__ RAM) {
    const int n    = blockIdx.z;
    const int it   = blockIdx.y;
    const int jt   = blockIdx.x;
    const int lane = threadIdx.x & 31;
    const int half = lane >> 4;
    const int mrow = lane & 15;

    const v2f* __restrict__ A =
        (const v2f*)(x1w + ((size_t)n * ITILES + it) * TILE_F) + lane;
    const v2f* __restrict__ B =
        (const v2f*)(x2w + ((size_t)n * JTILES + jt) * TILE_F) + lane;

    v8f acc = {};
    #pragma unroll
    for (int kc = 0; kc < 32; ++kc) {
        const v2f a = A[(size_t)kc * 32];
        const v2f b = B[(size_t)kc * 32];
        acc = __builtin_amdgcn_wmma_f32_16x16x4_f32(
            /*neg_a=*/false, a, /*neg_b=*/false, b,
            /*c_mod=*/(short)0, acc, /*reuse_a=*/false, /*reuse_b=*/false);
    }

    // D layout: VGPR r -> row it*16 + r + 8*half ; col = jt*16 + (lane&15)
    const int   j    = jt * 16 + mrow;
    const float s2j  = S2[n * TJ + j];
    const float invC = 1.f / (float)C_;
    #pragma unroll
    for (int r = 0; r < 8; ++r) {
        const int irow = it * 16 + r + 8 * half;
        if (irow < TI) {
            const float g = (S1[n * TIP + irow] + s2j - 2.f * acc[r]) * invC;
            float v = expf(g);
            if (v < 0.5f) v = 0.f;
            RAM[(((size_t)n * TIP) + irow) * TJ + j] = v;
        }
    }
}

// ---- kernel 4: column sums da[n,j] = sum_i RAM[n,i,j] ----
__global__ void k_colsum(const float* __restrict__ RAM, float* __restrict__ da) {
    const int j = threadIdx.x;   // 256 threads
    const int n = blockIdx.x;
    const float* base = RAM + ((size_t)n * TIP) * TJ + j;
    float s = 0.f;
    for (int i = 0; i < TI; ++i) s += base[(size_t)i * TJ];
    da[n * TJ + j] = s;
}

// ---- kernel 5: per-row db + normalized row sum -> w[n,i] ----
// w = 2 * sum_j RAM[i,j] / sqrt(db_i * da_j + eps)
__global__ void k_rownorm(const float* __restrict__ RAM, const float* __restrict__ da,
                          float* __restrict__ w) {
    const int i = blockIdx.x;    // 0..256
    const int n = blockIdx.y;
    const int j = threadIdx.x;   // 256 threads
    __shared__ float red[256];

    const float v = RAM[(((size_t)n * TIP) + i) * TJ + j];
    red[j] = v;
    __syncthreads();
    for (int s = 128; s > 0; s >>= 1) {
        if (j < s) red[j] += red[j + s];
        __syncthreads();
    }
    const float dbi = red[0];
    __syncthreads();

    const float contrib = v / sqrtf(dbi * da[n * TJ + j] + EPS_);
    red[j] = contrib;
    __syncthreads();
    for (int s = 128; s > 0; s >>= 1) {
        if (j < s) red[j] += red[j + s];
        __syncthreads();
    }
    if (j == 0) w[n * TIP + i] = 2.f * red[0];
}

// ---- kernel 6: out[n,c,t] = x[n,c,t] * w[n, t>>1] ----
__global__ void k_scale(const float* __restrict__ x, const float* __restrict__ w,
                        float* __restrict__ out) {
    const size_t idx   = (size_t)blockIdx.x * blockDim.x + threadIdx.x;
    const size_t total = (size_t)N_ * C_ * T_;
    if (idx >= total) return;
    const int t = (int)(idx % T_);
    const int n = (int)(idx / ((size_t)C_ * T_));
    out[idx] = x[idx] * w[n * TIP + (t >> 1)];
}

extern "C" void kernel_launch(void* const* d_in, const int* in_sizes, int n_in,
                              void* d_out, int out_size, void* d_ws, size_t ws_size,
                              hipStream_t stream) {
    const float* x   = (const float*)d_in[0];
    float*       out = (float*)d_out;
    float*       ws  = (float*)d_ws;

    // ws layout (floats): S1 | S2 | da | w | RAM | x1w | x2w   (~8.9 MB total)
    float* S1  = ws;                                // 16*272
    float* S2  = S1  + (size_t)N_ * TIP;            // 16*256
    float* da  = S2  + (size_t)N_ * TJ;             // 16*256
    float* w   = da  + (size_t)N_ * TJ;             // 16*272
    float* RAM = w   + (size_t)N_ * TIP;            // 16*272*256
    float* x1w = RAM + (size_t)N_ * TIP * TJ;       // 16*17*2048
    float* x2w = x1w + (size_t)N_ * ITILES * TILE_F; // 16*16*2048

    const size_t total = (size_t)N_ * C_ * T_;

    k_prep<<<dim3((T_ + 255) / 256, N_), 256, 0, stream>>>(x, S1, S2);
    k_pack<<<dim3((unsigned)((total + 255) / 256)), 256, 0, stream>>>(x, x1w, x2w);
    k_gemm<<<dim3(JTILES, ITILES, N_), 32, 0, stream>>>(x1w, x2w, S1, S2, RAM);
    k_colsum<<<dim3(N_), TJ, 0, stream>>>(RAM, da);
    k_rownorm<<<dim3(TI, N_), TJ, 0, stream>>>(RAM, da, w);
    k_scale<<<dim3((unsigned)((total + 255) / 256)), 256, 0, stream>>>(x, w, out);
}